// GCNBlockVer2_78314433675283
// MI455X (gfx1250) — compile-verified
//
#include <hip/hip_runtime.h>

typedef __attribute__((ext_vector_type(2))) float v2f;
typedef __attribute__((ext_vector_type(8))) float v8f;

#define N_NODES 16384
#define N_FEATS 512
#define D_NODE 128
#define D_EDGE 64
#define D_MSG 128
#define D_NODE_OUT 128
#define D_EDGE_OUT 64

__device__ __forceinline__ v8f wmma4(v2f a, v2f b, v8f c) {
  // V_WMMA_F32_16X16X4_F32 : D = A(16x4) * B(4x16) + C, full fp32
  return __builtin_amdgcn_wmma_f32_16x16x4_f32(
      /*neg_a=*/false, a, /*neg_b=*/false, b,
      /*c_mod=*/(short)0, c, /*reuse_a=*/false, /*reuse_b=*/false);
}

__device__ __forceinline__ void atomic_add_f32(float* p, float v) {
  // hardware global_atomic_add_f32 (no return), L2-resolved
  __hip_atomic_fetch_add(p, v, __ATOMIC_RELAXED, __HIP_MEMORY_SCOPE_AGENT);
}

// ---------------------------------------------------------------- utilities
__global__ void zero_kernel(float* __restrict__ p, long n) {
  long i = (long)blockIdx.x * blockDim.x + threadIdx.x;
  long stride = (long)gridDim.x * blockDim.x;
  for (; i < n; i += stride) p[i] = 0.0f;
}

// Rearrange row-major B (K x N) into WMMA B-operand order:
// packed[((k*NT + nt)*32 + lane)*2 + j] = B[4k + 2*(lane>>4) + j][nt*16 + (lane&15)]
__global__ void pack_kernel(const float* __restrict__ B, float* __restrict__ packed,
                            int K, int N) {
  int NT = N >> 4;
  int total = (K >> 2) * NT * 64;
  int i = blockIdx.x * blockDim.x + threadIdx.x;
  int stride = gridDim.x * blockDim.x;
  for (; i < total; i += stride) {
    int j = i & 1;
    int lane = (i >> 1) & 31;
    int nt = (i >> 6) % NT;
    int k = (i >> 6) / NT;
    int row = 4 * k + 2 * (lane >> 4) + j;
    int col = nt * 16 + (lane & 15);
    packed[i] = B[(size_t)row * N + col];
  }
}

// --------------------------------------------------- per-edge message GEMM
// msg_e = leaky_relu( [gatherEmb[gatherIdx[e]](128) , edgeEmb[e](64)] @ P + b )
// then atomic scatter-add into accum[scatterIdx[e]] and count into cnt.
// Two 16-edge M-subtiles per wave: each B fragment feeds 2 WMMAs.
__global__ void __launch_bounds__(256) edge_msg_kernel(
    const float* __restrict__ gatherEmb,   // (G, 128)
    const int* __restrict__ gatherIdx,     // (E)
    const float* __restrict__ edgeEmb,     // (E, 64)
    const float* __restrict__ packB,       // packed 192x128 (48 ksteps * 8 nt * 64)
    const float* __restrict__ bias,        // (128)
    const int* __restrict__ scatterIdx,    // (E)
    float* __restrict__ accum,             // (S, 128)
    float* __restrict__ cnt,               // (S)
    int nEdges) {
  int lane = threadIdx.x & 31;
  int wave = (int)((blockIdx.x * blockDim.x + threadIdx.x) >> 5);
  int nWaves = (int)((gridDim.x * blockDim.x) >> 5);
  int nTiles = (nEdges + 31) >> 5;  // 32-edge tiles
  int row = lane & 15;              // A row / C column-in-tile
  int kh = lane >> 4;               // K half select

  for (int t = wave; t < nTiles; t += nWaves) {
    int e0 = t << 5;
    const float* grow[2];
    const float* erow[2];
#pragma unroll
    for (int m = 0; m < 2; m++) {
      int e = e0 + m * 16 + row;
      if (e >= nEdges) e = nEdges - 1;
      int g = gatherIdx[e];
      grow[m] = gatherEmb + (size_t)g * 128;
      erow[m] = edgeEmb + (size_t)e * 64;
    }

    v8f acc[2][8];
#pragma unroll
    for (int nt = 0; nt < 8; nt++) {
      float bv = bias[nt * 16 + row];
      v8f bvv = {bv, bv, bv, bv, bv, bv, bv, bv};
      acc[0][nt] = bvv;
      acc[1][nt] = bvv;
    }

    // K-steps 0..31 : gathered embedding columns 0..127 (branch-uniform)
#pragma unroll 2
    for (int k = 0; k < 32; k++) {
      v2f a0 = *(const v2f*)(grow[0] + 4 * k + 2 * kh);
      v2f a1 = *(const v2f*)(grow[1] + 4 * k + 2 * kh);
      const float* bp = packB + (size_t)k * 512 + lane * 2;
#pragma unroll
      for (int nt = 0; nt < 8; nt++) {
        v2f b = *(const v2f*)(bp + nt * 64);
        acc[0][nt] = wmma4(a0, b, acc[0][nt]);
        acc[1][nt] = wmma4(a1, b, acc[1][nt]);
      }
    }
    // K-steps 32..47 : edge embedding columns 128..191
#pragma unroll 2
    for (int k = 32; k < 48; k++) {
      v2f a0 = *(const v2f*)(erow[0] + (4 * k - 128) + 2 * kh);
      v2f a1 = *(const v2f*)(erow[1] + (4 * k - 128) + 2 * kh);
      const float* bp = packB + (size_t)k * 512 + lane * 2;
#pragma unroll
      for (int nt = 0; nt < 8; nt++) {
        v2f b = *(const v2f*)(bp + nt * 64);
        acc[0][nt] = wmma4(a0, b, acc[0][nt]);
        acc[1][nt] = wmma4(a1, b, acc[1][nt]);
      }
    }

    // leaky relu + atomic scatter-mean numerator
#pragma unroll
    for (int m = 0; m < 2; m++) {
      int scat[8];
      int valid[8];
#pragma unroll
      for (int r = 0; r < 8; r++) {
        // C layout: lanes<16 -> rows 0..7, lanes>=16 -> rows 8..15
        int em = e0 + m * 16 + kh * 8 + r;
        valid[r] = (em < nEdges);
        scat[r] = scatterIdx[valid[r] ? em : (nEdges - 1)];
      }
#pragma unroll
      for (int nt = 0; nt < 8; nt++) {
        int c = nt * 16 + row;
#pragma unroll
        for (int r = 0; r < 8; r++) {
          if (valid[r]) {
            float x = acc[m][nt][r];
            x = x > 0.0f ? x : 0.01f * x;
            atomic_add_f32(accum + (size_t)scat[r] * 128 + c, x);
          }
        }
      }
    }
    int ec = e0 + lane;  // 32 lanes cover the 32 edges of this tile
    if (ec < nEdges) atomic_add_f32(cnt + scatterIdx[ec], 1.0f);
  }
}

// --------------------------------------------------------- scatter-mean div
__global__ void divide_kernel(float* __restrict__ accum, const float* __restrict__ cnt,
                              int rows) {
  int i = blockIdx.x * blockDim.x + threadIdx.x;
  int stride = gridDim.x * blockDim.x;
  int total = rows * 128;
  for (; i < total; i += stride) {
    float c = cnt[i >> 7];
    accum[i] = accum[i] / fmaxf(c, 1.0f);
  }
}

// ------------------------------------- out = emb@Q_self + msg@Q_neigh + b1+b2
// implemented as single K=256 WMMA GEMM; packB = [pack(Q_self); pack(Q_neigh)]
__global__ void __launch_bounds__(256) selfneigh_kernel(
    const float* __restrict__ emb,    // (R,128)
    const float* __restrict__ msg,    // (R,128)
    const float* __restrict__ packB,  // packed 256x128 (64 ksteps * 8 nt * 64)
    const float* __restrict__ bias1, const float* __restrict__ bias2,
    float* __restrict__ out,          // (R,128)
    int R) {
  int lane = threadIdx.x & 31;
  int wave = (int)((blockIdx.x * blockDim.x + threadIdx.x) >> 5);
  int nWaves = (int)((gridDim.x * blockDim.x) >> 5);
  int nTiles = (R + 15) >> 4;
  int row = lane & 15;
  int kh = lane >> 4;

  for (int t = wave; t < nTiles; t += nWaves) {
    int r0 = t << 4;
    int rr = r0 + row;
    if (rr >= R) rr = R - 1;
    const float* grow = emb + (size_t)rr * 128;
    const float* mrow = msg + (size_t)rr * 128;

    v8f acc[8];
#pragma unroll
    for (int nt = 0; nt < 8; nt++) {
      float bv = bias1[nt * 16 + row] + bias2[nt * 16 + row];
      v8f bvv = {bv, bv, bv, bv, bv, bv, bv, bv};
      acc[nt] = bvv;
    }
#pragma unroll 4
    for (int k = 0; k < 32; k++) {  // self half
      v2f a = *(const v2f*)(grow + 4 * k + 2 * kh);
      const float* bp = packB + (size_t)k * 512 + lane * 2;
#pragma unroll
      for (int nt = 0; nt < 8; nt++) {
        v2f b = *(const v2f*)(bp + nt * 64);
        acc[nt] = wmma4(a, b, acc[nt]);
      }
    }
#pragma unroll 4
    for (int k = 32; k < 64; k++) {  // neighbor-message half
      v2f a = *(const v2f*)(mrow + (4 * k - 128) + 2 * kh);
      const float* bp = packB + (size_t)k * 512 + lane * 2;
#pragma unroll
      for (int nt = 0; nt < 8; nt++) {
        v2f b = *(const v2f*)(bp + nt * 64);
        acc[nt] = wmma4(a, b, acc[nt]);
      }
    }
#pragma unroll
    for (int nt = 0; nt < 8; nt++) {
      int c = nt * 16 + row;
#pragma unroll
      for (int r = 0; r < 8; r++) {
        int m = r0 + kh * 8 + r;
        if (m < R) out[(size_t)m * 128 + c] = acc[nt][r];
      }
    }
  }
}

// ---------------- edge_out = [edge(64), node_out[src](128), feat_out[dst](128)] @ W + bW
// Two 16-edge M-subtiles per wave for B-fragment reuse.
__global__ void __launch_bounds__(256) edge_out_kernel(
    const float* __restrict__ edgeEmb,  // (E,64)
    const float* __restrict__ nodeOut,  // (N,128)
    const float* __restrict__ featOut,  // (F,128)
    const int* __restrict__ srcIdx, const int* __restrict__ dstIdx,
    const float* __restrict__ packW,    // packed 320x64 (80 ksteps * 4 nt * 64)
    const float* __restrict__ bW,       // (64)
    float* __restrict__ out,            // (E,64)
    int nEdges) {
  int lane = threadIdx.x & 31;
  int wave = (int)((blockIdx.x * blockDim.x + threadIdx.x) >> 5);
  int nWaves = (int)((gridDim.x * blockDim.x) >> 5);
  int nTiles = (nEdges + 31) >> 5;  // 32-edge tiles
  int row = lane & 15;
  int kh = lane >> 4;

  for (int t = wave; t < nTiles; t += nWaves) {
    int e0 = t << 5;
    const float* erow[2];
    const float* nrow[2];
    const float* frow[2];
#pragma unroll
    for (int m = 0; m < 2; m++) {
      int e = e0 + m * 16 + row;
      if (e >= nEdges) e = nEdges - 1;
      int s = srcIdx[e];
      int d = dstIdx[e];
      erow[m] = edgeEmb + (size_t)e * 64;
      nrow[m] = nodeOut + (size_t)s * 128;
      frow[m] = featOut + (size_t)d * 128;
    }

    v8f acc[2][4];
#pragma unroll
    for (int nt = 0; nt < 4; nt++) {
      float bv = bW[nt * 16 + row];
      v8f bvv = {bv, bv, bv, bv, bv, bv, bv, bv};
      acc[0][nt] = bvv;
      acc[1][nt] = bvv;
    }
#pragma unroll 4
    for (int k = 0; k < 16; k++) {  // edge segment, cols 0..63
      v2f a0 = *(const v2f*)(erow[0] + 4 * k + 2 * kh);
      v2f a1 = *(const v2f*)(erow[1] + 4 * k + 2 * kh);
      const float* bp = packW + (size_t)k * 256 + lane * 2;
#pragma unroll
      for (int nt = 0; nt < 4; nt++) {
        v2f b = *(const v2f*)(bp + nt * 64);
        acc[0][nt] = wmma4(a0, b, acc[0][nt]);
        acc[1][nt] = wmma4(a1, b, acc[1][nt]);
      }
    }
#pragma unroll 4
    for (int k = 16; k < 48; k++) {  // node_out segment, cols 64..191
      v2f a0 = *(const v2f*)(nrow[0] + (4 * k - 64) + 2 * kh);
      v2f a1 = *(const v2f*)(nrow[1] + (4 * k - 64) + 2 * kh);
      const float* bp = packW + (size_t)k * 256 + lane * 2;
#pragma unroll
      for (int nt = 0; nt < 4; nt++) {
        v2f b = *(const v2f*)(bp + nt * 64);
        acc[0][nt] = wmma4(a0, b, acc[0][nt]);
        acc[1][nt] = wmma4(a1, b, acc[1][nt]);
      }
    }
#pragma unroll 4
    for (int k = 48; k < 80; k++) {  // feat_out segment, cols 192..319
      v2f a0 = *(const v2f*)(frow[0] + (4 * k - 192) + 2 * kh);
      v2f a1 = *(const v2f*)(frow[1] + (4 * k - 192) + 2 * kh);
      const float* bp = packW + (size_t)k * 256 + lane * 2;
#pragma unroll
      for (int nt = 0; nt < 4; nt++) {
        v2f b = *(const v2f*)(bp + nt * 64);
        acc[0][nt] = wmma4(a0, b, acc[0][nt]);
        acc[1][nt] = wmma4(a1, b, acc[1][nt]);
      }
    }
#pragma unroll
    for (int m = 0; m < 2; m++) {
#pragma unroll
      for (int nt = 0; nt < 4; nt++) {
        int c = nt * 16 + row;
#pragma unroll
        for (int r = 0; r < 8; r++) {
          int em = e0 + m * 16 + kh * 8 + r;
          if (em < nEdges) out[(size_t)em * 64 + c] = acc[m][nt][r];
        }
      }
    }
  }
}

// ----------------------------------------------------------------- launcher
extern "C" void kernel_launch(void* const* d_in, const int* in_sizes, int n_in,
                              void* d_out, int out_size, void* d_ws, size_t ws_size,
                              hipStream_t stream) {
  const float* node_emb = (const float*)d_in[0];
  const float* edge_emb = (const float*)d_in[1];
  const float* feature_emb = (const float*)d_in[2];
  const int* src = (const int*)d_in[3];
  const int* dst = (const int*)d_in[4];
  const float* P_src = (const float*)d_in[5];
  const float* b_P_src = (const float*)d_in[6];
  const float* P_dst = (const float*)d_in[7];
  const float* b_P_dst = (const float*)d_in[8];
  const float* Q_src_self = (const float*)d_in[9];
  const float* b_Q_src_self = (const float*)d_in[10];
  const float* Q_src_neigh = (const float*)d_in[11];
  const float* b_Q_src_neigh = (const float*)d_in[12];
  const float* Q_dst_self = (const float*)d_in[13];
  const float* b_Q_dst_self = (const float*)d_in[14];
  const float* Q_dst_neigh = (const float*)d_in[15];
  const float* b_Q_dst_neigh = (const float*)d_in[16];
  const float* W = (const float*)d_in[17];
  const float* b_W = (const float*)d_in[18];

  int nEdges = in_sizes[3];

  // workspace layout (floats)
  float* ws = (float*)d_ws;
  float* accum_src = ws;                                     // 16384*128
  float* cnt_src = accum_src + (size_t)N_NODES * 128;        // 16384
  float* accum_dst = cnt_src + N_NODES;                      // 512*128
  float* cnt_dst = accum_dst + (size_t)N_FEATS * 128;        // 512
  float* packPsrc = cnt_dst + N_FEATS;                       // 24576
  float* packPdst = packPsrc + 24576;                        // 24576
  float* packQsrc = packPdst + 24576;                        // 32768
  float* packQdst = packQsrc + 32768;                        // 32768
  float* packWm = packQdst + 32768;                          // 20480

  // output layout: node_out (16384x128), edge_out (E x 64), feat_out (512x128)
  float* out_node = (float*)d_out;
  float* out_edge = out_node + (size_t)N_NODES * D_NODE_OUT;
  float* out_feat = out_edge + (size_t)nEdges * D_EDGE_OUT;

  long zeroN = (long)N_NODES * 128 + N_NODES + (long)N_FEATS * 128 + N_FEATS;
  zero_kernel<<<512, 256, 0, stream>>>(ws, zeroN);

  pack_kernel<<<64, 256, 0, stream>>>(P_src, packPsrc, 192, 128);
  pack_kernel<<<64, 256, 0, stream>>>(P_dst, packPdst, 192, 128);
  pack_kernel<<<64, 256, 0, stream>>>(Q_src_self, packQsrc, 128, 128);
  pack_kernel<<<64, 256, 0, stream>>>(Q_src_neigh, packQsrc + 16384, 128, 128);
  pack_kernel<<<64, 256, 0, stream>>>(Q_dst_self, packQdst, 128, 128);
  pack_kernel<<<64, 256, 0, stream>>>(Q_dst_neigh, packQdst + 16384, 128, 128);
  pack_kernel<<<64, 256, 0, stream>>>(W, packWm, 320, 64);

  // messages to nodes: gather feature_emb[dst], scatter-mean over src
  edge_msg_kernel<<<2048, 256, 0, stream>>>(feature_emb, dst, edge_emb, packPsrc,
                                            b_P_src, src, accum_src, cnt_src, nEdges);
  // messages to features: gather node_emb[src], scatter-mean over dst
  edge_msg_kernel<<<2048, 256, 0, stream>>>(node_emb, src, edge_emb, packPdst,
                                            b_P_dst, dst, accum_dst, cnt_dst, nEdges);

  divide_kernel<<<512, 256, 0, stream>>>(accum_src, cnt_src, N_NODES);
  divide_kernel<<<16, 256, 0, stream>>>(accum_dst, cnt_dst, N_FEATS);

  selfneigh_kernel<<<128, 256, 0, stream>>>(node_emb, accum_src, packQsrc,
                                            b_Q_src_self, b_Q_src_neigh, out_node,
                                            N_NODES);
  selfneigh_kernel<<<4, 256, 0, stream>>>(feature_emb, accum_dst, packQdst,
                                          b_Q_dst_self, b_Q_dst_neigh, out_feat,
                                          N_FEATS);

  edge_out_kernel<<<2048, 256, 0, stream>>>(edge_emb, out_node, out_feat, src, dst,
                                            packWm, b_W, out_edge, nEdges);
}